// RNN_4698694222213
// MI455X (gfx1250) — compile-verified
//
#include <hip/hip_runtime.h>
#include <hip/hip_bf16.h>

#define IC   196
#define NPTS (32 * 8192)

typedef float v2f __attribute__((ext_vector_type(2)));
typedef float v4f __attribute__((ext_vector_type(4)));
typedef float v8f __attribute__((ext_vector_type(8)));

// ---------------------------------------------------------------------------
// packA: 196 blocks x 128 floats. Lane l loads floats [4l..4l+3] with one
// global_load_b128 = {A_lo.v0, A_lo.v1, A_hi.v0, A_hi.v1}.
//
// A operand layout (16x4 f32): lane l<16 -> row l, v0=K0, v1=K1;
//                              lane l>=16 -> row l-16, v0=K2, v1=K3.
//
// A_lo (rows 0..5, serves points 0-15 via B's K0/K1 rows = lanes 0-15):
//     K0 = W1[i][0][o], K1 = W1[i][1][o], K2 = b1[i][o], K3 = 0
// A_hi (rows 8..13, serves points 16-31 via B's K2/K3 rows = lanes 16-31):
//     K0 = b1[i][o],   K1 = 0,            K2 = W1[i][0][o], K3 = W1[i][1][o]
// All other entries 0, so the two chained WMMAs fill disjoint halves of D
// and NO cross-lane data movement is ever needed for B.
//
// packB: 196 blocks x 8 floats: [W2[i][0..5], b2[i], 0] (SGPR broadcasts)
// Block 195 holds the final sigmoid cell (Wf1/bf1/Wf2/bf2).
// ---------------------------------------------------------------------------
__global__ void pack_weights(const float* __restrict__ W1, const float* __restrict__ b1,
                             const float* __restrict__ W2, const float* __restrict__ b2,
                             const float* __restrict__ Wf1, const float* __restrict__ bf1,
                             const float* __restrict__ Wf2, const float* __restrict__ bf2,
                             float* __restrict__ packA, float* __restrict__ packB) {
  const int i = blockIdx.x;       // step 0..195
  const int t = threadIdx.x;      // 0..127
  const bool last = (i == IC - 1);
  const float* w1 = last ? Wf1 : (W1 + i * 12);   // [2][6] row-major
  const float* bb = last ? bf1 : (b1 + i * 6);

  const int l = t >> 2;           // destination lane
  const int c = t & 3;            // 0=A_lo.v0, 1=A_lo.v1, 2=A_hi.v0, 3=A_hi.v1
  float v = 0.0f;
  switch (c) {
    case 0:  // A_lo.v0 : rows 0..5 K0 = W1[0] | rows 0..5 K2 = b1
      if (l < 6)                     v = w1[l];
      else if (l >= 16 && l < 22)    v = bb[l - 16];
      break;
    case 1:  // A_lo.v1 : rows 0..5 K1 = W1[1]   (K3 = 0)
      if (l < 6)                     v = w1[6 + l];
      break;
    case 2:  // A_hi.v0 : rows 8..13 K0 = b1 | rows 8..13 K2 = W1[0]
      if (l >= 8 && l < 14)          v = bb[l - 8];
      else if (l >= 24 && l < 30)    v = w1[l - 24];
      break;
    case 3:  // A_hi.v1 : rows 8..13 K3 = W1[1]  (K1 = 0)
      if (l >= 24 && l < 30)         v = w1[6 + (l - 24)];
      break;
  }
  packA[i * 128 + t] = v;

  if (t < 8) {
    float pv = 0.0f;
    if (t < 6)       pv = last ? Wf2[t] : W2[i * 6 + t];
    else if (t == 6) pv = last ? bf2[0] : b2[i];
    packB[i * 8 + t] = pv;
  }
}

// ---------------------------------------------------------------------------
// One wave = 32 points (lane = point). Per step, two chained WMMAs fill one D:
//   d = A_lo x {lo? x :1, y} + 0   -> D rows 0..5  <- points 0-15 (K0/K1)
//   d = A_hi x {lo? 1 :x, y} + d   -> D rows 8..13 <- points 16-31 (K2/K3)
// d[0..5] then holds h-channel o for ALL 32 lanes; layer 2 runs full-lane:
// 6x relu + 6 fma against SGPR-held W2/b2. y stays in natural lanes forever.
// ---------------------------------------------------------------------------
__global__ __launch_bounds__(256) void rnn_wmma(const float* __restrict__ x,
                                                const float* __restrict__ packA,
                                                const float* __restrict__ packB,
                                                float* __restrict__ out) {
  const int lane = threadIdx.x & 31;
  const int wave = blockIdx.x * (blockDim.x >> 5) + (threadIdx.x >> 5);
  const int base = wave * 32;
  const int p    = base + lane;

  const float* xr   = x + (size_t)p * IC;   // 784B stride -> float4 aligned
  const float* arow = packA + lane * 4;     // per-lane A fetch (b128)
  const bool   lo   = (lane < 16);

  float y = xr[0];                          // y0 = x[..., 0] (all 32 points)

  auto cell = [&](int i, float xi) -> float {
    v4f a = *(const v4f*)(arow + (size_t)i * 128);   // one global_load_b128
    v2f alo; alo.x = a.x; alo.y = a.y;
    v2f ahi; ahi.x = a.z; ahi.y = a.w;

    v2f b0, b1v;
    b0.x  = lo ? xi : 1.0f;   b0.y  = y;    // K0 = x (pts 0-15) | K2-bias row
    b1v.x = lo ? 1.0f : xi;   b1v.y = y;    // K0-bias row | K2 = x (pts 16-31)

    v8f c = {};
    v8f d = __builtin_amdgcn_wmma_f32_16x16x4_f32(false, alo, false, b0,
                                                  (short)0, c, false, false);
    d     = __builtin_amdgcn_wmma_f32_16x16x4_f32(false, ahi, false, b1v,
                                                  (short)0, d, false, false);

    const float* wb = packB + i * 8;        // uniform -> s_load
    float acc = wb[6];                      // b2
#pragma unroll
    for (int o = 0; o < 6; ++o)
      acc = fmaf(fmaxf(d[o], 0.0f), wb[o], acc);   // relu(h).W2 + b2, 32 lanes
    return acc;
  };

#pragma unroll 1
  for (int cch = 0; cch < 48; ++cch) {      // steps 0..191
    const float4 xv = *(const float4*)(xr + 4 * cch);
    y = fmaxf(cell(4 * cch + 0, xv.x), 0.0f);
    y = fmaxf(cell(4 * cch + 1, xv.y), 0.0f);
    y = fmaxf(cell(4 * cch + 2, xv.z), 0.0f);
    y = fmaxf(cell(4 * cch + 3, xv.w), 0.0f);
  }
  {                                         // steps 192..195 (last = sigmoid)
    const float4 xv = *(const float4*)(xr + 192);
    y = fmaxf(cell(192, xv.x), 0.0f);
    y = fmaxf(cell(193, xv.y), 0.0f);
    y = fmaxf(cell(194, xv.z), 0.0f);
    const float acc = cell(195, xv.w);
    y = 1.0f / (1.0f + __expf(-acc));
  }

  out[p] = y;                               // all 32 lanes carry results
}

extern "C" void kernel_launch(void* const* d_in, const int* in_sizes, int n_in,
                              void* d_out, int out_size, void* d_ws, size_t ws_size,
                              hipStream_t stream) {
  const float* x   = (const float*)d_in[0];
  const float* W1  = (const float*)d_in[1];
  const float* b1  = (const float*)d_in[2];
  const float* W2  = (const float*)d_in[3];
  const float* b2  = (const float*)d_in[4];
  const float* Wf1 = (const float*)d_in[5];
  const float* bf1 = (const float*)d_in[6];
  const float* Wf2 = (const float*)d_in[7];
  const float* bf2 = (const float*)d_in[8];

  float* packA = (float*)d_ws;              // 196 * 128 floats = 100352 B
  float* packB = packA + IC * 128;          // 196 *   8 floats =   6272 B

  pack_weights<<<IC, 128, 0, stream>>>(W1, b1, W2, b2, Wf1, bf1, Wf2, bf2, packA, packB);

  const int waves = NPTS / 32;              // 8192 waves, 32 points each
  const int block = 256;                    // 8 waves / block
  const int grid  = waves / 8;              // 1024 blocks
  rnn_wmma<<<grid, block, 0, stream>>>(x, packA, packB, (float*)d_out);
}